// TransitionPlaceEmbeddingModel_77979426226636
// MI455X (gfx1250) — compile-verified
//
#include <hip/hip_runtime.h>
#include <cmath>

typedef __attribute__((ext_vector_type(2))) float v2f;
typedef __attribute__((ext_vector_type(8))) float v8f;

#define NB   128   // batch
#define LSEQ 128   // sequence length
#define TNUM 32    // transitions
#define DIMV 64
#define K4V  68    // DIM + 4
#define MR   144   // 129 rows padded to 9*16
#define HSTR 132   // LDS stride for 128-wide buffers (bank-conflict padding)
#define WSTR 66    // LDS stride for 64-wide buffer
#define NEG_SLOPE 0.01f
#define BETAV 0.01f

__device__ __forceinline__ float leakyf(float x) { return x > 0.f ? x : NEG_SLOPE * x; }
__device__ __forceinline__ float reluf(float x)  { return x > 0.f ? x : 0.f; }

// Native fp32 WMMA: D(16x16) = A(16x4) * B(4x16) + C, wave32.
__device__ __forceinline__ v8f wmma_f32x4(v2f a, v2f b, v8f c) {
  return __builtin_amdgcn_wmma_f32_16x16x4_f32(false, a, false, b, (short)0, c, false, false);
}

// ---------------------------------------------------------------------------
// Kernel 1: the sequential scan. One block per batch element; marking lives in
// LDS for all 128 steps; BOTH weight matrices live in registers for all steps
// (loop-invariant B-fragments hoisted out of the scan), so the steady state is
// LDS + WMMA plus per-step global reads of pcW1's cond rows, biases and inputs.
// ---------------------------------------------------------------------------
__global__ __launch_bounds__(256) void scan_kernel(
    const float* __restrict__ bd,   const float* __restrict__ G,
    const float* __restrict__ Cm,   const float* __restrict__ Temb,
    const float* __restrict__ pcW1, const float* __restrict__ pcb1,
    const float* __restrict__ pcW2, const float* __restrict__ pcb2,
    float* __restrict__ wsMark)
{
  __shared__ float smMark[MR * K4V];   // marking, stride 68
  __shared__ float smH[MR * HSTR];     // layer-1 activations
  __shared__ float smW[MR * WSTR];     // layer-2 logits / softmax weights
  __shared__ float sCondC[DIMV];
  __shared__ float sCondT[4];
  __shared__ float sBias1[128];
  __shared__ float sZm[MR];
  __shared__ int   sActive, sAi, sGi;

  const int b    = blockIdx.x;
  const int tid  = threadIdx.x;
  const int wave = tid >> 5;
  const int lane = tid & 31;
  const int half = lane >> 4;   // selects K offset 0 or 2 in fp32 WMMA fragments
  const int l16  = lane & 15;

  // ---- hoist loop-invariant B-fragments into registers (held all 128 steps)
  // GEMM1: each wave owns N-tile `wave` of pcW1[:68] (68x128)
  const int col1 = wave * 16 + l16;
  v2f bW1[17];
  #pragma unroll
  for (int ks = 0; ks < 17; ++ks) {
    const int k0 = ks * 4 + half * 2;
    bW1[ks].x = pcW1[k0 * 128 + col1];
    bW1[ks].y = pcW1[(k0 + 1) * 128 + col1];
  }
  // GEMM2: each wave owns fixed N-tile (wave&3) of pcW2 (128x64);
  // waves 0-3 cover mt 0..4, waves 4-7 cover mt 5..8 (all 36 tiles).
  const int nt2    = wave & 3;
  const int col2   = nt2 * 16 + l16;
  const int mtBeg2 = (wave >> 2) * 5;          // 0 or 5
  const int mtEnd2 = (wave >> 2) ? 9 : 5;
  v2f bW2[32];
  #pragma unroll
  for (int ks = 0; ks < 32; ++ks) {
    const int k0 = ks * 4 + half * 2;
    bW2[ks].x = pcW2[k0 * DIMV + col2];
    bW2[ks].y = pcW2[(k0 + 1) * DIMV + col2];
  }

  // init marking: all zero, row 0 = relu(G[0]) in cols [0,64)
  for (int idx = tid; idx < MR * K4V; idx += 256) smMark[idx] = 0.f;
  __syncthreads();
  if (tid < DIMV) smMark[tid] = reluf(G[tid]);

  for (int i = 0; i < LSEQ; ++i) {
    __syncthreads();
    if (tid == 0) {
      int a = (int)bd[b * 5 * LSEQ + i];
      sActive = (a > 0 && a != TNUM + 1) ? 1 : 0;
      int ai = a - 1; if (ai < 0) ai = 0; if (ai > TNUM - 1) ai = TNUM - 1;
      int gi = a;     if (gi < 0) gi = 0; if (gi > TNUM)     gi = TNUM;
      sAi = ai; sGi = gi;
    }
    __syncthreads();
    if (!sActive) continue;                    // uniform across block

    if (tid < DIMV)               sCondC[tid]        = reluf(Cm[sAi * DIMV + tid]);
    else if (tid < DIMV + 4)      sCondT[tid - DIMV] = Temb[sAi * 4 + (tid - DIMV)];
    __syncthreads();

    // effective layer-1 bias: pcb1 + cond-part of x @ pcW1 (same for all rows)
    if (tid < 128) {
      float acc = pcb1[tid];
      for (int k = 0; k < DIMV; ++k) acc += sCondC[k] * pcW1[(K4V + k) * 128 + tid];
      for (int k = 0; k < 4; ++k)    acc += sCondT[k] * pcW1[(K4V + DIMV + k) * 128 + tid];
      sBias1[tid] = acc;
    }
    // zero-row mask on pre-update marking
    for (int r = tid; r < MR; r += 256) {
      int z = 1;
      for (int c = 0; c < K4V; ++c) z &= (smMark[r * K4V + c] == 0.f);
      sZm[r] = (float)z;
    }
    __syncthreads();

    // GEMM1: h = leaky(mark(144x68) @ pcW1[:68](68x128) + bias1)
    for (int mt = 0; mt < 9; ++mt) {
      v8f acc = {};
      const int row = mt * 16 + l16;
      #pragma unroll
      for (int ks = 0; ks < 17; ++ks) {
        const int k0 = ks * 4 + half * 2;
        v2f a; a.x = smMark[row * K4V + k0]; a.y = smMark[row * K4V + k0 + 1];
        acc = wmma_f32x4(a, bW1[ks], acc);
      }
      const int drow = mt * 16 + half * 8;
      #pragma unroll
      for (int r = 0; r < 8; ++r)
        smH[(drow + r) * HSTR + col1] = leakyf(acc[r] + sBias1[col1]);
    }
    __syncthreads();

    // GEMM2: w = h(144x128) @ pcW2(128x64) + pcb2   (B in registers)
    for (int mt = mtBeg2; mt < mtEnd2; ++mt) {
      const int row = mt * 16 + l16;
      v8f acc = {};
      #pragma unroll
      for (int ks = 0; ks < 32; ++ks) {
        const int k0 = ks * 4 + half * 2;
        v2f a; a.x = smH[row * HSTR + k0]; a.y = smH[row * HSTR + k0 + 1];
        acc = wmma_f32x4(a, bW2[ks], acc);
      }
      const int drow = mt * 16 + half * 8;
      #pragma unroll
      for (int r = 0; r < 8; ++r)
        smW[(drow + r) * WSTR + col2] = acc[r] + pcb2[col2];
    }
    __syncthreads();

    // masked softmax over the 129 positions, one column per thread
    if (tid < DIMV) {
      float m = -3.0e38f;
      for (int r = 0; r <= LSEQ; ++r) {
        float v = (sZm[r] != 0.f) ? -1.0e9f : smW[r * WSTR + tid];
        if (v > m) m = v;
      }
      float s = 0.f;
      for (int r = 0; r <= LSEQ; ++r) {
        float v = (sZm[r] != 0.f) ? -1.0e9f : smW[r * WSTR + tid];
        float e = expf(v - m);
        smW[r * WSTR + tid] = e;
        s += e;
      }
      const float inv = 1.f / s;
      for (int r = 0; r <= LSEQ; ++r) smW[r * WSTR + tid] *= inv;
    }
    __syncthreads();

    // consume: new_place = relu(mark[:, :64] - w * condC)
    for (int idx = tid; idx < (LSEQ + 1) * DIMV; idx += 256) {
      const int r = idx >> 6, c = idx & 63;
      smMark[r * K4V + c] = reluf(smMark[r * K4V + c] - smW[r * WSTR + c] * sCondC[c]);
    }
    __syncthreads();

    // write generated row i+1 = [relu(G[gi]), t]
    if (tid < K4V) {
      float v;
      if (tid < DIMV) v = reluf(G[sGi * DIMV + tid]);
      else            v = bd[b * 5 * LSEQ + (1 + (tid - DIMV)) * LSEQ + i];
      smMark[(i + 1) * K4V + tid] = v;
    }
  }
  __syncthreads();
  // final marking -> workspace (rows 0..128 are contiguous at stride 68)
  for (int idx = tid; idx < (LSEQ + 1) * K4V; idx += 256)
    wsMark[b * (LSEQ + 1) * K4V + idx] = smMark[idx];
}

// ---------------------------------------------------------------------------
// Kernel 2: per-(b,t) scoring. Two 129x68x128 WMMA GEMMs reusing one LDS buf.
// ---------------------------------------------------------------------------
__global__ __launch_bounds__(256) void score_kernel(
    const float* __restrict__ wsMark,
    const float* __restrict__ jcW1, const float* __restrict__ jcb1,
    const float* __restrict__ jfW,  const float* __restrict__ jfb,
    const float* __restrict__ jtW1, const float* __restrict__ jtb1,
    const float* __restrict__ jtW2, const float* __restrict__ jtb2,
    float* __restrict__ out)
{
  __shared__ float smMark[MR * K4V];
  __shared__ float smWgt[K4V * HSTR];
  __shared__ float smBuf[MR * HSTR];
  __shared__ float sZm[MR];
  __shared__ float sPool[128];
  __shared__ float sTc[MR];
  __shared__ float sJt2[128];
  __shared__ float sRes;

  const int t    = blockIdx.x;
  const int b    = blockIdx.y;
  const int tid  = threadIdx.x;
  const int wave = tid >> 5;
  const int lane = tid & 31;
  const int half = lane >> 4;
  const int l16  = lane & 15;

  for (int idx = tid; idx < MR * K4V; idx += 256)
    smMark[idx] = (idx < (LSEQ + 1) * K4V) ? wsMark[b * (LSEQ + 1) * K4V + idx] : 0.f;
  for (int idx = tid; idx < K4V * 128; idx += 256) {
    const int k = idx >> 7, n = idx & 127;
    smWgt[k * HSTR + n] = jcW1[(t * K4V + k) * 128 + n];
  }
  __syncthreads();
  for (int r = tid; r < MR; r += 256) {
    int z = 1;
    for (int c = 0; c < K4V; ++c) z &= (smMark[r * K4V + c] == 0.f);
    sZm[r] = (float)z;
  }
  __syncthreads();

  // Phase A: cond = leaky(mark @ jcW1[t] + jcb1[t]), masked rows -> 0
  {
    const int col = wave * 16 + l16;
    const float bias = jcb1[t * 128 + col];
    v2f bfr[17];
    #pragma unroll
    for (int ks = 0; ks < 17; ++ks) {
      const int k0 = ks * 4 + half * 2;
      bfr[ks].x = smWgt[k0 * HSTR + col];
      bfr[ks].y = smWgt[(k0 + 1) * HSTR + col];
    }
    for (int mt = 0; mt < 9; ++mt) {
      v8f acc = {};
      const int row = mt * 16 + l16;
      #pragma unroll
      for (int ks = 0; ks < 17; ++ks) {
        const int k0 = ks * 4 + half * 2;
        v2f a; a.x = smMark[row * K4V + k0]; a.y = smMark[row * K4V + k0 + 1];
        acc = wmma_f32x4(a, bfr[ks], acc);
      }
      const int drow = mt * 16 + half * 8;
      #pragma unroll
      for (int r = 0; r < 8; ++r) {
        const int orow = drow + r;
        float v = leakyf(acc[r] + bias);
        if (sZm[orow] != 0.f) v = 0.f;
        smBuf[orow * HSTR + col] = v;
      }
    }
  }
  __syncthreads();

  // pooled sum over positions; meanwhile stage jt weights
  if (tid < 128) {
    float s = 0.f;
    for (int r = 0; r <= LSEQ; ++r) s += smBuf[r * HSTR + tid];
    sPool[tid] = s;
  }
  for (int idx = tid; idx < K4V * 128; idx += 256) {
    const int k = idx >> 7, n = idx & 127;
    smWgt[k * HSTR + n] = jtW1[(t * K4V + k) * 128 + n];
  }
  if (tid < 128) sJt2[tid] = jtW2[t * 128 + tid];
  __syncthreads();

  // Phase B: th = leaky(mark @ jtW1[t] + jtb1[t]) (no mask)
  {
    const int col = wave * 16 + l16;
    const float bias = jtb1[t * 128 + col];
    v2f bfr[17];
    #pragma unroll
    for (int ks = 0; ks < 17; ++ks) {
      const int k0 = ks * 4 + half * 2;
      bfr[ks].x = smWgt[k0 * HSTR + col];
      bfr[ks].y = smWgt[(k0 + 1) * HSTR + col];
    }
    for (int mt = 0; mt < 9; ++mt) {
      v8f acc = {};
      const int row = mt * 16 + l16;
      #pragma unroll
      for (int ks = 0; ks < 17; ++ks) {
        const int k0 = ks * 4 + half * 2;
        v2f a; a.x = smMark[row * K4V + k0]; a.y = smMark[row * K4V + k0 + 1];
        acc = wmma_f32x4(a, bfr[ks], acc);
      }
      const int drow = mt * 16 + half * 8;
      #pragma unroll
      for (int r = 0; r < 8; ++r)
        smBuf[(drow + r) * HSTR + col] = leakyf(acc[r] + bias);
    }
  }
  __syncthreads();

  if (tid == 0) {
    float d = jfb[0];
    for (int k = 0; k < 128; ++k) d += sPool[k] * jfW[k];
    float res = 1.f / (1.f + expf(-d));
    sRes = fmaxf(res, 1e-5f);
  }
  for (int r = tid; r <= LSEQ; r += 256) {
    float s = jtb2[t];
    for (int k = 0; k < 128; ++k) s += smBuf[r * HSTR + k] * sJt2[k];
    sTc[r] = s;
  }
  __syncthreads();

  if (tid == 0) {
    float m = -3.0e38f;
    for (int r = 0; r <= LSEQ; ++r) m = fmaxf(m, sTc[r]);
    const float res = sRes;
    out[b * TNUM + t] = res * m - BETAV * expf(res);
  }
}

// ---------------------------------------------------------------------------
extern "C" void kernel_launch(void* const* d_in, const int* in_sizes, int n_in,
                              void* d_out, int out_size, void* d_ws, size_t ws_size,
                              hipStream_t stream) {
  const float* bd   = (const float*)d_in[0];
  const float* G    = (const float*)d_in[1];
  const float* Cm   = (const float*)d_in[2];
  const float* Temb = (const float*)d_in[3];
  const float* pcW1 = (const float*)d_in[4];
  const float* pcb1 = (const float*)d_in[5];
  const float* pcW2 = (const float*)d_in[6];
  const float* pcb2 = (const float*)d_in[7];
  const float* jcW1 = (const float*)d_in[8];
  const float* jcb1 = (const float*)d_in[9];
  const float* jfW  = (const float*)d_in[10];
  const float* jfb  = (const float*)d_in[11];
  const float* jtW1 = (const float*)d_in[12];
  const float* jtb1 = (const float*)d_in[13];
  const float* jtW2 = (const float*)d_in[14];
  const float* jtb2 = (const float*)d_in[15];

  float* wsMark = (float*)d_ws;   // B * 129 * 68 floats = ~4.5 MB
  float* out    = (float*)d_out;  // (B, TN)

  scan_kernel<<<dim3(NB), dim3(256), 0, stream>>>(
      bd, G, Cm, Temb, pcW1, pcb1, pcW2, pcb2, wsMark);
  score_kernel<<<dim3(TNUM, NB), dim3(256), 0, stream>>>(
      wsMark, jcW1, jcb1, jfW, jfb, jtW1, jtb1, jtW2, jtb2, out);
}